// DCR_62294205661990
// MI455X (gfx1250) — compile-verified
//
#include <hip/hip_runtime.h>
#include <math.h>

typedef float v2f __attribute__((ext_vector_type(2)));
typedef float v8f __attribute__((ext_vector_type(8)));

#define MAX_ANS_LEN 30
#define EPS_F 1e-8f
#define S_MAX 2048
#define P2_THREADS 256

// ---------------------------------------------------------------------------
// Order-preserving float <-> uint encoding for atomicMax-based scatter-max.
// ---------------------------------------------------------------------------
__device__ __forceinline__ unsigned enc_f(float x) {
  unsigned u = __float_as_uint(x);
  return (u & 0x80000000u) ? ~u : (u | 0x80000000u);
}
__device__ __forceinline__ float dec_f(unsigned u) {
  u = (u & 0x80000000u) ? (u & 0x7FFFFFFFu) : ~u;
  return __uint_as_float(u);
}

// ---------------------------------------------------------------------------
// Phase 1: one pass over seq. Per wave: a 16-row tile of one example.
//   WMMA F32 16x16x4:  A = seq tile (16xK), B cols {0,1} = {q1,q2} chunk.
//   D[:,0] = dot1 rows, D[:,1] = dot2 rows.  nsq accumulated in VALU.
// A layout (32-bit 16x4): lane L<16 -> M=L, holds K=kbase+{0,1};
//                         lane L>=16 -> M=L-16, holds K=kbase+{2,3}.
// B layout (32-bit 4x16): lane stripes N=lane&15, VGPR0/1 hold K-halves.
// C/D layout: VGPR r, lanes 0-15 -> (M=r, N=lane); lanes 16-31 -> (M=8+r).
// ---------------------------------------------------------------------------
__global__ __launch_bounds__(128)
void dcr_dot_pass(const float* __restrict__ seq, const int* __restrict__ idxs,
                  float* __restrict__ dot1, float* __restrict__ dot2,
                  float* __restrict__ nsq, int S, int H) {
  const int wave = threadIdx.x >> 5;
  const int lane = threadIdx.x & 31;
  const int tiles_per_blk = blockDim.x >> 5;          // 4 waves/block
  const int tiles_per_ex  = S >> 4;                   // S/16 tiles per example
  const int blocks_per_ex = tiles_per_ex / tiles_per_blk;
  const int b    = blockIdx.x / blocks_per_ex;
  const int tile = (blockIdx.x % blocks_per_ex) * tiles_per_blk + wave;
  const int row_base = tile << 4;

  const int sep0 = idxs[2 * b];
  const float* base = seq + (size_t)b * (size_t)S * (size_t)H;
  const float* q1 = base + (size_t)H;                   // seq[b,1,:]
  const float* q2 = base + (size_t)(sep0 - 1) * (size_t)H;

  const int m  = lane & 15;   // row within tile (A) / column N (B)
  const int kh = lane >> 4;   // K-half: 0 -> K{0,1}, 1 -> K{2,3}
  const float* rowp = base + (size_t)(row_base + m) * (size_t)H + kh * 2;
  const float* qp   = ((m == 0) ? q1 : q2) + kh * 2;
  const bool hasq   = (m < 2);

  v8f acc0 = {0.f, 0.f, 0.f, 0.f, 0.f, 0.f, 0.f, 0.f};
  v8f acc1 = {0.f, 0.f, 0.f, 0.f, 0.f, 0.f, 0.f, 0.f};
  const v2f zero2 = {0.f, 0.f};
  float npart = 0.f;

#pragma unroll 2
  for (int k = 0; k < H; k += 8) {
    // chunk 0: K = k..k+3
    v2f a0 = *(const v2f*)(rowp + k);
    v2f bq0 = *(const v2f*)(qp + k);          // safe address for all lanes
    v2f b0 = hasq ? bq0 : zero2;
    acc0 = __builtin_amdgcn_wmma_f32_16x16x4_f32(
        false, a0, false, b0, (short)0, acc0, false, false);
    npart = fmaf(a0.x, a0.x, npart);
    npart = fmaf(a0.y, a0.y, npart);

    // chunk 1: K = k+4..k+7 (independent accumulator breaks WMMA RAW chain)
    v2f a1 = *(const v2f*)(rowp + k + 4);
    v2f bq1 = *(const v2f*)(qp + k + 4);
    v2f b1 = hasq ? bq1 : zero2;
    acc1 = __builtin_amdgcn_wmma_f32_16x16x4_f32(
        false, a1, false, b1, (short)0, acc1, false, false);
    npart = fmaf(a1.x, a1.x, npart);
    npart = fmaf(a1.y, a1.y, npart);
  }

  // Row nsq: lanes L and L+16 jointly hold row m's elements.
  float nrow = npart + __shfl_xor(npart, 16, 32);
  const size_t off = (size_t)b * (size_t)S;
  if (lane < 16) nsq[off + row_base + m] = nrow;

  // Extract dot1 (column 0) and dot2 (column 1) from D.
#pragma unroll
  for (int r = 0; r < 8; ++r) {
    float v = acc0[r] + acc1[r];
    int row = row_base + (kh << 3) + r;
    if (m == 0)      dot1[off + row] = v;
    else if (m == 1) dot2[off + row] = v;
  }
}

// ---------------------------------------------------------------------------
// Phase 2: banded cosine-sim window, scatter-max, stats, sign flip.
// One block per example; everything staged in LDS.
// ---------------------------------------------------------------------------
__device__ __forceinline__ float block_reduce(float v, float* buf, int t,
                                              bool is_max) {
  buf[t] = v;
  __syncthreads();
  for (int s = P2_THREADS / 2; s > 0; s >>= 1) {
    if (t < s) buf[t] = is_max ? fmaxf(buf[t], buf[t + s]) : (buf[t] + buf[t + s]);
    __syncthreads();
  }
  float r = buf[0];
  __syncthreads();
  return r;
}

__global__ __launch_bounds__(P2_THREADS)
void dcr_band_pass(const float* __restrict__ dot1, const float* __restrict__ dot2,
                   const float* __restrict__ nsq, const int* __restrict__ idxs,
                   float* __restrict__ out, int B, int S) {
  __shared__ float s_d1[S_MAX];
  __shared__ float s_d2[S_MAX];
  __shared__ float s_n[S_MAX];
  __shared__ unsigned s_end[S_MAX];
  __shared__ float s_start[S_MAX];
  __shared__ float s_endl[S_MAX];
  __shared__ float s_red[P2_THREADS];

  const int b = blockIdx.x;
  const int t = threadIdx.x;
  const int sep0 = idxs[2 * b];
  const int sep1 = idxs[2 * b + 1];
  const size_t off = (size_t)b * (size_t)S;
  const float NEG_INF = -__builtin_inff();
  const unsigned NEG_INF_ENC = enc_f(NEG_INF);

  for (int i = t; i < S; i += P2_THREADS) {
    s_d1[i] = dot1[off + i];
    s_d2[i] = dot2[off + i];
    s_n[i]  = nsq[off + i];
    s_end[i] = NEG_INF_ENC;
  }
  __syncthreads();

  const float qnorm = sqrtf(s_n[1] + s_n[sep0 - 1]);
  const float qden  = fmaxf(qnorm, EPS_F);

  for (int i = t; i < S; i += P2_THREADS) {
    const float d1 = s_d1[i];
    const float ni = s_n[i];
    const bool iok = (i >= sep0 + 1) && (i < sep1);
    float best = NEG_INF;
    int arg = 0;
    bool anyv = false;
#pragma unroll
    for (int d = 0; d < MAX_ANS_LEN; ++d) {
      int jr = i + d;
      int j = min(jr, S - 1);
      float num = d1 + s_d2[j];
      float den = fmaxf(sqrtf(ni + s_n[j]), EPS_F) * qden;
      float sim = num / den;
      bool v = iok && (jr < sep1);
      float sm = v ? sim : NEG_INF;
      if (sm > best) { best = sm; arg = d; }   // strict '>' -> first max wins
      anyv |= v;
    }
    s_start[i] = anyv ? best : 0.0f;
    if (anyv) {
      int ei = min(i + arg, S - 1);
      atomicMax(&s_end[ei], enc_f(best));
    }
  }
  __syncthreads();

  for (int i = t; i < S; i += P2_THREADS) {
    float e = dec_f(s_end[i]);
    s_endl[i] = (e == NEG_INF) ? 0.0f : e;
  }
  __syncthreads();

  float pmax = NEG_INF, pss = 0.f, pss2 = 0.f, pes = 0.f, pes2 = 0.f;
  for (int i = t; i < S; i += P2_THREADS) {
    float x = s_start[i], y = s_endl[i];
    pmax = fmaxf(pmax, x);
    pss += x; pss2 += x * x;
    pes += y; pes2 += y * y;
  }
  float maxl  = block_reduce(pmax, s_red, t, true);
  float sums  = block_reduce(pss,  s_red, t, false);
  float sums2 = block_reduce(pss2, s_red, t, false);
  float sume  = block_reduce(pes,  s_red, t, false);
  float sume2 = block_reduce(pes2, s_red, t, false);

  const float fS = (float)S;
  float ms = sums / fS;
  float ss = sqrtf(fmaxf(0.f, sums2 - fS * ms * ms) / (fS - 1.f));
  float me = sume / fS;
  float se = sqrtf(fmaxf(0.f, sume2 - fS * me * me) / (fS - 1.f));
  bool flip = (maxl < ms + ss) || (maxl < me + se);

  float* out_s = out + off;
  float* out_e = out + (size_t)B * (size_t)S + off;
  for (int i = t; i < S; i += P2_THREADS) {
    float x = s_start[i], y = s_endl[i];
    if (flip) {
      x = (x == 0.0f) ? -0.001f : -x;
      y = (y == 0.0f) ? -0.001f : -y;
    }
    out_s[i] = x;
    out_e[i] = y;
  }
}

// ---------------------------------------------------------------------------
extern "C" void kernel_launch(void* const* d_in, const int* in_sizes, int n_in,
                              void* d_out, int out_size, void* d_ws, size_t ws_size,
                              hipStream_t stream) {
  const float* seq = (const float*)d_in[0];
  const int* idxs  = (const int*)d_in[1];
  float* out = (float*)d_out;

  const int B = in_sizes[1] / 2;                       // idxs is [B,2]
  const int S = out_size / (2 * B);                    // outputs: 2 x [B,S]
  const long long tot = (long long)in_sizes[0];
  const int H = (int)(tot / ((long long)B * (long long)S));

  float* dot1 = (float*)d_ws;
  float* dot2 = dot1 + (size_t)B * (size_t)S;
  float* nsqb = dot2 + (size_t)B * (size_t)S;

  const int tiles_per_ex  = S / 16;
  const int blocks_per_ex = tiles_per_ex / 4;          // 4 waves per block
  dcr_dot_pass<<<dim3(B * blocks_per_ex), dim3(128), 0, stream>>>(
      seq, idxs, dot1, dot2, nsqb, S, H);
  dcr_band_pass<<<dim3(B), dim3(P2_THREADS), 0, stream>>>(
      dot1, dot2, nsqb, idxs, out, B, S);
}